// ProductAttributeModel_47785806135987
// MI455X (gfx1250) — compile-verified
//
#include <hip/hip_runtime.h>
#include <math.h>

// ---------------------------------------------------------------------------
// ProductAttributeModel on MI455X (gfx1250, wave32, WMMA).
// Routed (sparse) expert computation instead of the reference's dense-all-
// categories compute; fp32 WMMA 16x16x4 with f32 accumulate for numerical
// fidelity to the fp32 reference.
// ---------------------------------------------------------------------------

typedef __attribute__((ext_vector_type(2))) float v2f;
typedef __attribute__((ext_vector_type(8))) float v8f;

#define NB     1024   // batch
#define NF     2048   // input features
#define NH1    1024
#define NH2    512
#define NCAT   8
#define NATT   5
#define NCMAX  32
#define TOTPAD 1152   // >= 1024 + 8*15, rounded up

__device__ __forceinline__ float gelu_exact(float x) {
  // torch nn.GELU default: exact erf formulation
  return 0.5f * x * (1.0f + erff(x * 0.7071067811865476f));
}

__device__ __forceinline__ v8f wmma_f32(v2f a, v2f b, v8f c) {
  // D = A(16x4,f32) * B(4x16,f32) + C(16x16,f32)
  return __builtin_amdgcn_wmma_f32_16x16x4_f32(
      /*neg_a=*/false, a, /*neg_b=*/false, b,
      /*c_mod=*/(short)0, c, /*reuse_a=*/false, /*reuse_b=*/false);
}

// ---------------------------------------------------------------------------
// Routing: counting sort of batch rows by category into 16-padded segments.
// meta layout (ints): [0..7] counts, [8..16] padded offsets (off[8]=total),
//                     [32..32+TOTPAD) perm (source row, or -1 for pad slot)
// ---------------------------------------------------------------------------
__global__ __launch_bounds__(256) void route_kernel(
    const int* __restrict__ cats, int* __restrict__ meta) {
  __shared__ int cnt[NCAT];
  __shared__ int base[NCAT + 1];
  __shared__ int cur[NCAT];
  const int tid = threadIdx.x;
  if (tid < NCAT) { cnt[tid] = 0; cur[tid] = 0; }
  __syncthreads();
  for (int b = tid; b < NB; b += 256) atomicAdd(&cnt[cats[b]], 1);
  __syncthreads();
  if (tid == 0) {
    int off = 0;
    for (int c = 0; c < NCAT; c++) {
      base[c] = off;
      off += (cnt[c] + 15) & ~15;   // pad each segment to a 16-row tile
    }
    base[NCAT] = off;
  }
  __syncthreads();
  const int tot = base[NCAT];
  for (int i = tid; i < tot; i += 256) meta[32 + i] = -1;
  __syncthreads();
  for (int b = tid; b < NB; b += 256) {
    int c = cats[b];
    int p = atomicAdd(&cur[c], 1);
    meta[32 + base[c] + p] = b;
  }
  if (tid < NCAT) meta[tid] = cnt[tid];
  if (tid <= NCAT) meta[8 + tid] = base[tid];
}

// ---------------------------------------------------------------------------
// Fused GEMM(+bias)+GELU+LayerNorm tile core.
// One 256-thread block (8 wave32s) computes 16 output rows x N cols so the
// LayerNorm row reduction stays block-local. Each wave owns CT = N/128
// 16x16 WMMA accumulator tiles. X tile staged in LDS (stride 33 avoids bank
// conflicts); epilogue keeps activations in registers and reduces mean/var
// through a small LDS tree (two-pass, biased variance, eps=1e-5, matching
// the reference exactly).
// ---------------------------------------------------------------------------
template <int N, int K, int CT>
__device__ __forceinline__ void mlp_tile_core(
    const float* __restrict__ X, long xstride,
    const int* __restrict__ perm, int rowbase,
    const float* __restrict__ W,      // K x N
    const float* __restrict__ bias,   // N
    const float* __restrict__ g,      // N (LN gamma)
    const float* __restrict__ be,     // N (LN beta)
    float* __restrict__ Y, long ystride) {
  constexpr int KC = 32;             // K chunk staged in LDS
  __shared__ float xs[16 * 33];      // 16 x KC, padded stride
  __shared__ float red[16 * 128];    // per-(wave,lane) row partials
  __shared__ float red2[16 * 16];
  __shared__ float stat[16];         // mean per row
  __shared__ float stat2[16];        // rstd per row

  const int tid  = threadIdx.x;
  const int wave = tid >> 5;
  const int lane = tid & 31;
  const int half = lane >> 4;        // 0: K{0,1}/rows 0-7; 1: K{2,3}/rows 8-15
  const int l    = lane & 15;
  const int col0 = wave * (CT * 16);

  const v8f vzero = {0.f, 0.f, 0.f, 0.f, 0.f, 0.f, 0.f, 0.f};
  v8f acc[CT];
#pragma unroll
  for (int t = 0; t < CT; t++) acc[t] = vzero;

  for (int kc = 0; kc < K; kc += KC) {
    // Stage 16 x KC tile of X (with optional row gather through perm).
#pragma unroll
    for (int i = tid; i < 16 * KC; i += 256) {
      int r = i >> 5;
      int k = i & 31;
      int pr = rowbase + r;
      int src = perm ? perm[pr] : pr;
      xs[r * 33 + k] = (src >= 0) ? X[(long)src * xstride + kc + k] : 0.0f;
    }
    __syncthreads();
#pragma unroll
    for (int k = 0; k < KC; k += 4) {
      v2f af;                                   // A fragment: row l, K pair
      af.x = xs[l * 33 + k + 2 * half];
      af.y = xs[l * 33 + k + 2 * half + 1];
      const float* wrow = W + (long)(kc + k + 2 * half) * N + col0 + l;
#pragma unroll
      for (int t = 0; t < CT; t++) {
        v2f bf;                                 // B fragment: col l, K pair
        bf.x = wrow[t * 16];
        bf.y = wrow[t * 16 + N];
        acc[t] = wmma_f32(af, bf, acc[t]);
      }
    }
    __syncthreads();
  }

  // bias + exact GELU, in registers
#pragma unroll
  for (int t = 0; t < CT; t++) {
    float bv = bias[col0 + t * 16 + l];
#pragma unroll
    for (int i = 0; i < 8; i++) acc[t][i] = gelu_exact(acc[t][i] + bv);
  }

  // ---- LayerNorm pass 1: mean ----
#pragma unroll
  for (int i = 0; i < 8; i++) {
    float s = 0.f;
#pragma unroll
    for (int t = 0; t < CT; t++) s += acc[t][i];
    red[(i + 8 * half) * 128 + wave * 16 + l] = s;
  }
  __syncthreads();
  {
    int r = tid >> 4, sub = tid & 15;
    float s = 0.f;
#pragma unroll
    for (int j = 0; j < 8; j++) s += red[r * 128 + sub + j * 16];
    red2[r * 16 + sub] = s;
  }
  __syncthreads();
  if (tid < 16) {
    float s = 0.f;
    for (int j = 0; j < 16; j++) s += red2[tid * 16 + j];
    stat[tid] = s / (float)N;
  }
  __syncthreads();

  // ---- LayerNorm pass 2: biased variance ----
#pragma unroll
  for (int i = 0; i < 8; i++) {
    float m = stat[i + 8 * half];
    float s = 0.f;
#pragma unroll
    for (int t = 0; t < CT; t++) {
      float d = acc[t][i] - m;
      s += d * d;
    }
    red[(i + 8 * half) * 128 + wave * 16 + l] = s;
  }
  __syncthreads();
  {
    int r = tid >> 4, sub = tid & 15;
    float s = 0.f;
#pragma unroll
    for (int j = 0; j < 8; j++) s += red[r * 128 + sub + j * 16];
    red2[r * 16 + sub] = s;
  }
  __syncthreads();
  if (tid < 16) {
    float s = 0.f;
    for (int j = 0; j < 16; j++) s += red2[tid * 16 + j];
    stat2[tid] = 1.0f / sqrtf(s / (float)N + 1e-5f);
  }
  __syncthreads();

  // ---- normalize + affine, write out ----
#pragma unroll
  for (int t = 0; t < CT; t++) {
    int col = col0 + t * 16 + l;
    float gg = g[col], bb = be[col];
#pragma unroll
    for (int i = 0; i < 8; i++) {
      int r = i + 8 * half;
      float val = (acc[t][i] - stat[r]) * stat2[r] * gg + bb;
      Y[(long)(rowbase + r) * ystride + col] = val;
    }
  }
}

// Dense (shared-weight) layer: feature processor.
template <int N, int K, int CT>
__global__ __launch_bounds__(256) void dense_kernel(
    const float* __restrict__ X, long xstride,
    const float* __restrict__ W, const float* __restrict__ bias,
    const float* __restrict__ g, const float* __restrict__ be,
    float* __restrict__ Y, long ystride) {
  mlp_tile_core<N, K, CT>(X, xstride, nullptr, (int)blockIdx.x * 16,
                          W, bias, g, be, Y, ystride);
}

// Routed expert layer: per-(category, attr) weights; grid = (tiles, A, C).
template <int N, int K, int CT, bool GATHER>
__global__ __launch_bounds__(256) void expert_kernel(
    const float* __restrict__ Xb, long xstride, long x_a_off,
    const float* __restrict__ Wb, const float* __restrict__ bb,
    const float* __restrict__ gb, const float* __restrict__ beb,
    float* __restrict__ Yb, long ystride, long y_a_off,
    const int* __restrict__ meta) {
  const int c = blockIdx.z, a = blockIdx.y;
  const int off = meta[8 + c];
  const int tiles = (meta[9 + c] - off) >> 4;   // padded rows / 16
  if ((int)blockIdx.x >= tiles) return;         // uniform exit: EXEC stays full
  const long e = (long)(c * NATT + a);
  mlp_tile_core<N, K, CT>(
      Xb + (long)a * x_a_off, xstride,
      GATHER ? (meta + 32) : nullptr,
      off + (int)blockIdx.x * 16,
      Wb + e * (long)K * N, bb + e * N, gb + e * N, beb + e * N,
      Yb + (long)a * y_a_off, ystride);
}

// Final head: 16x32 logits tile, bias, NC-mask, scatter to out[b, a, :].
__global__ __launch_bounds__(64) void logits_kernel(
    const float* __restrict__ h2, const float* __restrict__ hw3,
    const float* __restrict__ hb3, const int* __restrict__ meta,
    float* __restrict__ out) {
  const int c = blockIdx.z, a = blockIdx.y;
  const int off = meta[8 + c];
  const int tiles = (meta[9 + c] - off) >> 4;
  if ((int)blockIdx.x >= tiles) return;
  const int rowbase = off + (int)blockIdx.x * 16;
  const int* perm = meta + 32;

  const int tid  = threadIdx.x;
  const int wave = tid >> 5;
  const int lane = tid & 31;
  const int half = lane >> 4;
  const int l    = lane & 15;
  const int col0 = wave * 16;                 // wave0: cols 0-15, wave1: 16-31

  const long e = (long)(c * NATT + a);
  const float* X    = h2 + (long)a * 256;     // row stride NATT*256
  const float* W    = hw3 + e * 256 * NCMAX;
  const float* bias = hb3 + e * NCMAX;

  const v8f vzero = {0.f, 0.f, 0.f, 0.f, 0.f, 0.f, 0.f, 0.f};
  v8f acc = vzero;
  const long xrow = (long)(rowbase + l) * (NATT * 256);
  for (int k = 0; k < 256; k += 4) {
    v2f af;
    af.x = X[xrow + k + 2 * half];
    af.y = X[xrow + k + 2 * half + 1];
    const float* wrow = W + (long)(k + 2 * half) * NCMAX + col0 + l;
    v2f bf;
    bf.x = wrow[0];
    bf.y = wrow[NCMAX];
    acc = wmma_f32(af, bf, acc);
  }

  const int nc = 8 + (7 * c + 11 * a) % 25;   // NC table (host constant)
  const int col = col0 + l;
  const float bv = bias[col];
  const float neginf = -__builtin_inff();
#pragma unroll
  for (int i = 0; i < 8; i++) {
    int r = i + 8 * half;
    int b = perm[rowbase + r];
    if (b >= 0) {                              // skip pad rows
      float val = acc[i] + bv;
      out[((long)b * NATT + a) * NCMAX + col] = (col < nc) ? val : neginf;
    }
  }
}

// ---------------------------------------------------------------------------
extern "C" void kernel_launch(void* const* d_in, const int* in_sizes, int n_in,
                              void* d_out, int out_size, void* d_ws, size_t ws_size,
                              hipStream_t stream) {
  const float* x    = (const float*)d_in[0];
  const int*   cats = (const int*)d_in[1];
  // d_in[2] = n_attrs scalar (== NATT), compile-time constant here
  const float* pw1  = (const float*)d_in[3];
  const float* pb1  = (const float*)d_in[4];
  const float* pg1  = (const float*)d_in[5];
  const float* pbe1 = (const float*)d_in[6];
  const float* pw2  = (const float*)d_in[7];
  const float* pb2  = (const float*)d_in[8];
  const float* pg2  = (const float*)d_in[9];
  const float* pbe2 = (const float*)d_in[10];
  const float* hw1  = (const float*)d_in[11];
  const float* hb1  = (const float*)d_in[12];
  const float* hg1  = (const float*)d_in[13];
  const float* hbe1 = (const float*)d_in[14];
  const float* hw2  = (const float*)d_in[15];
  const float* hb2  = (const float*)d_in[16];
  const float* hg2  = (const float*)d_in[17];
  const float* hbe2 = (const float*)d_in[18];
  const float* hw3  = (const float*)d_in[19];
  const float* hb3  = (const float*)d_in[20];
  float* out = (float*)d_out;

  // Workspace layout
  int*   meta = (int*)d_ws;                                  // 8 KB
  float* f1 = (float*)((char*)d_ws + 8192);                  // B x H1
  float* f2 = f1 + (size_t)NB * NH1;                         // B x H2
  float* h1 = f2 + (size_t)NB * NH2;                         // TOTPAD x A x 512
  float* h2 = h1 + (size_t)TOTPAD * NATT * 512;              // TOTPAD x A x 256

  route_kernel<<<1, 256, 0, stream>>>(cats, meta);

  // Feature processor: 1024x2048x1024 and 1024x1024x512
  dense_kernel<NH1, NF, 8><<<NB / 16, 256, 0, stream>>>(
      x, NF, pw1, pb1, pg1, pbe1, f1, NH1);
  dense_kernel<NH2, NH1, 4><<<NB / 16, 256, 0, stream>>>(
      f1, NH1, pw2, pb2, pg2, pbe2, f2, NH2);

  // Routed expert heads (only the owned category per row)
  dim3 ge(NB / 16, NATT, NCAT);
  expert_kernel<512, NH2, 4, true><<<ge, 256, 0, stream>>>(
      f2, NH2, /*x_a_off=*/0, hw1, hb1, hg1, hbe1,
      h1, /*ystride=*/(long)NATT * 512, /*y_a_off=*/512, meta);
  expert_kernel<256, 512, 2, false><<<ge, 256, 0, stream>>>(
      h1, /*xstride=*/(long)NATT * 512, /*x_a_off=*/512, hw2, hb2, hg2, hbe2,
      h2, /*ystride=*/(long)NATT * 256, /*y_a_off=*/256, meta);

  logits_kernel<<<ge, 64, 0, stream>>>(h2, hw3, hb3, meta, out);
}